// MeanAggregator_1400159339187
// MI455X (gfx1250) — compile-verified
//
#include <hip/hip_runtime.h>

#define N_NODES 10000
#define D_FEAT  128
#define BATCH   8192
#define KNBR    25

#define WAVES_PER_BLOCK 4
#define LANES 32

// ---------------------------------------------------------------------------
// CDNA5 async global->LDS gather (GLOBAL_LOAD_ASYNC_TO_LDS_B128, ASYNCcnt).
// Per-lane: copies 16 bytes from the lane's global address to the lane's LDS
// address. 32 lanes x 16B = one 512B feature row per issue.
// ---------------------------------------------------------------------------
typedef int v4i __attribute__((ext_vector_type(4)));
typedef __attribute__((address_space(1))) v4i* gv4_t;   // global AS pointer
typedef __attribute__((address_space(3))) v4i* lv4_t;   // LDS AS pointer

__device__ __forceinline__ void async_copy_b128(const float* gsrc, float* ldst) {
#if __has_builtin(__builtin_amdgcn_global_load_async_to_lds_b128)
    __builtin_amdgcn_global_load_async_to_lds_b128(
        (gv4_t)(v4i*)(float*)gsrc,   // cast away const, reinterpret, AS1
        (lv4_t)(v4i*)ldst,           // reinterpret, AS3
        /*offset=*/0, /*cpol=*/0);
#else
    unsigned loff = (unsigned)(size_t)(__attribute__((address_space(3))) void*)ldst;
    asm volatile("global_load_async_to_lds_b128 %0, %1, off"
                 :: "v"(loff), "v"(gsrc) : "memory");
#endif
}

__device__ __forceinline__ void wait_async_zero() {
#if __has_builtin(__builtin_amdgcn_s_wait_asynccnt)
    __builtin_amdgcn_s_wait_asynccnt(0);
#else
    asm volatile("s_wait_asynccnt 0" ::: "memory");
#endif
}

__device__ __forceinline__ float readlane_f32(float v, int lane) {
    return __int_as_float(__builtin_amdgcn_readlane(__float_as_int(v), lane));
}

// ---------------------------------------------------------------------------
// One wave32 per batch row.
// ---------------------------------------------------------------------------
__global__ __launch_bounds__(WAVES_PER_BLOCK * LANES)
void mean_agg_kernel(const int*   __restrict__ nodes,      // [BATCH]
                     const int*   __restrict__ neigh_idx,  // [BATCH,K]
                     const float* __restrict__ feat,       // [N,128]
                     const float* __restrict__ adj,        // [N,N]
                     const float* __restrict__ sims,       // [N,N]
                     float*       __restrict__ out)        // [BATCH,128]
{
    __shared__ float lds_feat[WAVES_PER_BLOCK][KNBR * D_FEAT]; // 4 * 12.8KB

    const int lane = threadIdx.x & (LANES - 1);
    const int wave = threadIdx.x >> 5;
    // wave-uniform batch row index, forced into an SGPR
    const int b = __builtin_amdgcn_readfirstlane(blockIdx.x * WAVES_PER_BLOCK + wave);

    // ---- scalar row id (s_load) and per-lane neighbor gather (lanes 0..24) ----
    const int row = nodes[b];
    const float* __restrict__ adjRow = adj  + (size_t)row * N_NODES;
    const float* __restrict__ simRow = sims + (size_t)row * N_NODES;

    int   idx = 0;
    float w   = 0.0f;
    if (lane < KNBR) {
        idx = neigh_idx[(size_t)b * KNBR + lane];
        w   = adjRow[idx] + simRow[idx];   // scattered 4B HBM gathers
    }

    // ---- wave32 reduction for denom ----
    float denom = w;
    #pragma unroll
    for (int s = 16; s > 0; s >>= 1)
        denom += __shfl_xor(denom, s, LANES);

    // ---- async-gather the 25 feature rows into LDS (saddr + lane offset) ----
    float* lbase = &lds_feat[wave][0];
    #pragma unroll
    for (int j = 0; j < KNBR; ++j) {
        const int ij = __builtin_amdgcn_readlane(idx, j);   // SGPR row id
        const float* g = feat + (size_t)ij * D_FEAT + lane * 4;
        float*       l = lbase + j * D_FEAT + lane * 4;
        async_copy_b128(g, l);                              // 16B per lane
    }
    wait_async_zero();                                      // ASYNCcnt == 0

    // ---- weighted accumulation: each lane owns 4 features ----
    const int d = lane * 4;
    float4 acc = make_float4(0.f, 0.f, 0.f, 0.f);
    #pragma unroll
    for (int j = 0; j < KNBR; ++j) {
        const float wj = readlane_f32(w, j);                // SGPR weight
        const float4 v = *(const float4*)(lbase + j * D_FEAT + d); // ds_load_b128
        acc.x = fmaf(wj, v.x, acc.x);
        acc.y = fmaf(wj, v.y, acc.y);
        acc.z = fmaf(wj, v.z, acc.z);
        acc.w = fmaf(wj, v.w, acc.w);
    }

    const float inv = 1.0f / denom;
    float4 res = make_float4(acc.x * inv, acc.y * inv, acc.z * inv, acc.w * inv);
    *(float4*)(out + (size_t)b * D_FEAT + d) = res;         // global_store_b128
}

extern "C" void kernel_launch(void* const* d_in, const int* in_sizes, int n_in,
                              void* d_out, int out_size, void* d_ws, size_t ws_size,
                              hipStream_t stream) {
    const int*   nodes = (const int*)  d_in[0];
    const int*   neigh = (const int*)  d_in[1];
    const float* feat  = (const float*)d_in[2];
    const float* adj   = (const float*)d_in[3];
    const float* sims  = (const float*)d_in[4];
    float*       out   = (float*)      d_out;

    const int blocks = BATCH / WAVES_PER_BLOCK; // 2048
    mean_agg_kernel<<<blocks, WAVES_PER_BLOCK * LANES, 0, stream>>>(
        nodes, neigh, feat, adj, sims, out);
}